// CRFMRC_89988154786458
// MI455X (gfx1250) — compile-verified
//
#include <hip/hip_runtime.h>
#include <math.h>

#define BSZ   64
#define TLEN  512
#define DDIM  768
#define LNUM  5
#define NSEQ  128            // 2*B
#define NROW  (NSEQ * TLEN)  // 65536 rows of x / logits

typedef __attribute__((ext_vector_type(2))) float v2f;
typedef __attribute__((ext_vector_type(8))) float v8f;

// ---------------------------------------------------------------------------
// Kernel 1: logits[row][n] = x[row][:] . W[:][n] + b[n]   (n padded 5 -> 16)
// One wave32 per 16-row tile, V_WMMA_F32_16X16X4_F32 over K=768 (192 steps).
// A layout (16x4 f32): lane = (h<<4)|m ; VGPR0 = X[m][k0+2h], VGPR1 = X[m][k0+2h+1]
// B layout (4x16 f32): lane = (h<<4)|n ; VGPR0 = W[k0+2h][n], VGPR1 = W[k0+2h+1][n]
// C/D layout: VGPR r, lane (h<<4)|n  ->  D[r + 8h][n]
// ---------------------------------------------------------------------------
__global__ __launch_bounds__(256) void gemm_logits_wmma(
    const float* __restrict__ X, const float* __restrict__ W,
    const float* __restrict__ bias, float* __restrict__ out) {
  __shared__ v2f ldsb[192 * 32];  // 48KB: per-frag, per-lane B operand
  const int tid = threadIdx.x;

  // Cooperative precompute of zero-padded B fragments (shared by all 8 waves).
  for (int idx = tid; idx < 192 * 32; idx += 256) {
    const int f = idx >> 5, l = idx & 31;
    const int h = l >> 4, n = l & 15;
    const int k = 4 * f + 2 * h;
    v2f v;
    v.x = (n < LNUM) ? W[k * LNUM + n] : 0.0f;
    v.y = (n < LNUM) ? W[(k + 1) * LNUM + n] : 0.0f;
    ldsb[idx] = v;
  }
  __syncthreads();

  const int lane = tid & 31;
  const int wave = tid >> 5;
  const int h = lane >> 4;
  const int mn = lane & 15;  // M for A loads, N for B/C/D
  const int rowBase = blockIdx.x * 128 + wave * 16;

  const float* xptr = X + (size_t)(rowBase + mn) * DDIM + 2 * h;

  v8f acc = {0.f, 0.f, 0.f, 0.f, 0.f, 0.f, 0.f, 0.f};
#pragma unroll 4
  for (int f = 0; f < 192; ++f) {
    v2f a = *(const v2f*)(xptr + 4 * f);   // global_load_b64, 8B aligned
    v2f bf = ldsb[f * 32 + lane];          // ds_load_b64
    acc = __builtin_amdgcn_wmma_f32_16x16x4_f32(
        /*neg_a=*/false, a, /*neg_b=*/false, bf,
        /*c_mod=*/(short)0, acc, /*reuse_a=*/false, /*reuse_b=*/false);
  }

  if (mn < LNUM) {
    const float bn = bias[mn];
#pragma unroll
    for (int r = 0; r < 8; ++r) {
      const int row = rowBase + r + 8 * h;
      out[(size_t)row * LNUM + mn] = acc[r] + bn;
    }
  }
}

// ---------------------------------------------------------------------------
// Kernel 2: per-sequence CRF: log-norm scan + Viterbi fwd/backtrace + score.
// One wave32 per sequence; lanes 0..4 carry the 5 states; alpha broadcast
// via wave shuffles; backpointers in LDS (512x5 bytes).
// ---------------------------------------------------------------------------
__global__ __launch_bounds__(32) void crf_forward(
    const float* __restrict__ logits, const int* __restrict__ label,
    const int* __restrict__ seqlen, const float* __restrict__ trans,
    int* __restrict__ tags, float* __restrict__ ll) {
  const int nseq = blockIdx.x;   // 0..127
  const int lane = threadIdx.x;  // 0..31
  const int b0 = nseq & 63;      // original batch index
  const int half = nseq >> 6;    // 0 -> label1, 1 -> label2
  const int sl = seqlen[b0];

  __shared__ float str[32];                 // 5x5 transitions
  __shared__ unsigned char bp[TLEN * LNUM]; // backpointers
  if (lane < 25) str[lane] = trans[lane];
  __syncthreads();

  const int j = (lane < LNUM) ? lane : 0;  // this lane's destination state
  float tr[5];
#pragma unroll
  for (int i = 0; i < 5; ++i) tr[i] = str[i * 5 + j];

  float alpha[5], valpha[5];
#pragma unroll
  for (int i = 0; i < 5; ++i) {
    const float v = logits[((size_t)nseq * TLEN) * LNUM + i];
    alpha[i] = v;
    valpha[i] = v;
  }

  for (int t = 1; t < TLEN; ++t) {
    const float xt = logits[((size_t)nseq * TLEN + t) * LNUM + j];
    // log-sum-exp over sources i
    float mm = alpha[0] + tr[0];
#pragma unroll
    for (int i = 1; i < 5; ++i) mm = fmaxf(mm, alpha[i] + tr[i]);
    float ss = 0.f;
#pragma unroll
    for (int i = 0; i < 5; ++i) ss += __expf(alpha[i] + tr[i] - mm);
    const float na_new = xt + mm + __logf(ss);
    // viterbi max / first-argmax over sources i
    float vm = valpha[0] + tr[0];
    int am = 0;
#pragma unroll
    for (int i = 1; i < 5; ++i) {
      const float s = valpha[i] + tr[i];
      if (s > vm) { vm = s; am = i; }
    }
    const bool keep = (t < sl);
    const float na = keep ? na_new : alpha[j];
    const float nv = keep ? (xt + vm) : valpha[j];
    const int nb = keep ? am : j;
    if (lane < LNUM) bp[t * LNUM + lane] = (unsigned char)nb;
#pragma unroll
    for (int i = 0; i < 5; ++i) {
      alpha[i] = __shfl(na, i, 32);
      valpha[i] = __shfl(nv, i, 32);
    }
  }
  __syncthreads();

  // log-norm (identical in every lane)
  float mm = alpha[0];
#pragma unroll
  for (int i = 1; i < 5; ++i) mm = fmaxf(mm, alpha[i]);
  float ss = 0.f;
#pragma unroll
  for (int i = 0; i < 5; ++i) ss += __expf(alpha[i] - mm);
  const float lognorm = mm + __logf(ss);

  // last tag = first argmax of valpha
  float vm = valpha[0];
  int ltag = 0;
#pragma unroll
  for (int i = 1; i < 5; ++i)
    if (valpha[i] > vm) { vm = valpha[i]; ltag = i; }

  // serial backtrace (lane 0)
  if (lane == 0) {
    int tg = ltag;
    tags[(size_t)nseq * TLEN + (TLEN - 1)] = tg;
    for (int t = TLEN - 1; t >= 1; --t) {
      tg = bp[t * LNUM + tg];
      tags[(size_t)nseq * TLEN + t - 1] = tg;
    }
  }

  // gold-path sequence score (lane-strided over t, shuffle reduction)
  float sc = 0.f;
  for (int t = lane; t < TLEN; t += 32) {
    if (t < sl) {
      const int lt = label[b0 * TLEN + t];
      const int tg = half ? (lt > 0 ? lt + 1 : lt) : (lt > 0 ? 1 : lt);
      sc += logits[((size_t)nseq * TLEN + t) * LNUM + tg];
      if (t >= 1) {
        const int lp = label[b0 * TLEN + t - 1];
        const int tp = half ? (lp > 0 ? lp + 1 : lp) : (lp > 0 ? 1 : lp);
        sc += str[tp * 5 + tg];
      }
    }
  }
#pragma unroll
  for (int off = 16; off > 0; off >>= 1) sc += __shfl_down(sc, off, 32);
  if (lane == 0) ll[nseq] = sc - lognorm;
}

// ---------------------------------------------------------------------------
// Kernel 3: viterbi = (v1==0 | v2==0) ? 0 : v2-1, written as float to d_out.
// ---------------------------------------------------------------------------
__global__ __launch_bounds__(256) void combine_viterbi(
    const int* __restrict__ tags, float* __restrict__ out) {
  const int idx = blockIdx.x * 256 + threadIdx.x;  // idx = b*TLEN + t
  if (idx >= BSZ * TLEN) return;
  const int v1 = tags[idx];
  const int v2 = tags[(size_t)BSZ * TLEN + idx];
  out[idx] = (float)((v1 == 0 || v2 == 0) ? 0 : (v2 - 1));
}

// ---------------------------------------------------------------------------
// Kernel 4: loss = mean(-ll), acc = sum((pred==label)*mask)/sum(seqlen).
// ---------------------------------------------------------------------------
__global__ __launch_bounds__(256) void finalize(
    const float* __restrict__ outvit, const int* __restrict__ label,
    const int* __restrict__ seqlen, const float* __restrict__ ll,
    float* __restrict__ out) {
  __shared__ float r0[256], r1[256], r2[256];
  const int tid = threadIdx.x;

  float lsum = 0.f;
  for (int n = tid; n < NSEQ; n += 256) lsum -= ll[n];

  float corr = 0.f;
  for (int idx = tid; idx < BSZ * TLEN; idx += 256) {
    const int b = idx >> 9;          // / TLEN
    const int t = idx & (TLEN - 1);  // % TLEN
    if (t < seqlen[b]) {
      const int pred = (int)outvit[idx];
      corr += (pred == label[idx]) ? 1.f : 0.f;
    }
  }

  float slen = 0.f;
  for (int b = tid; b < BSZ; b += 256) slen += (float)seqlen[b];

  r0[tid] = lsum; r1[tid] = corr; r2[tid] = slen;
  __syncthreads();
  for (int s = 128; s > 0; s >>= 1) {
    if (tid < s) {
      r0[tid] += r0[tid + s];
      r1[tid] += r1[tid + s];
      r2[tid] += r2[tid + s];
    }
    __syncthreads();
  }
  if (tid == 0) {
    out[BSZ * TLEN] = r0[0] / (float)NSEQ;  // loss
    out[BSZ * TLEN + 1] = r1[0] / r2[0];    // acc
  }
}

// ---------------------------------------------------------------------------
extern "C" void kernel_launch(void* const* d_in, const int* in_sizes, int n_in,
                              void* d_out, int out_size, void* d_ws,
                              size_t ws_size, hipStream_t stream) {
  (void)in_sizes; (void)n_in; (void)out_size; (void)ws_size;
  const float* X      = (const float*)d_in[0];  // (128, 512, 768) f32
  const int*   label  = (const int*)d_in[1];    // (64, 512) i32
  const int*   seqlen = (const int*)d_in[2];    // (64,) i32
  const float* W      = (const float*)d_in[3];  // (768, 5) f32
  const float* bias   = (const float*)d_in[4];  // (5,) f32
  const float* trans  = (const float*)d_in[5];  // (5, 5) f32
  float* out = (float*)d_out;                   // 64*512 viterbi + loss + acc

  char* ws = (char*)d_ws;
  float* logits = (float*)ws;                              // 128*512*5 f32 = 1310720 B
  int*   tags   = (int*)(ws + 1310720);                    // 128*512 i32  =  262144 B
  float* ll     = (float*)(ws + 1310720 + 262144);         // 128 f32

  gemm_logits_wmma<<<NROW / 128, 256, 0, stream>>>(X, W, bias, logits);
  crf_forward<<<NSEQ, 32, 0, stream>>>(logits, label, seqlen, trans, tags, ll);
  combine_viterbi<<<(BSZ * TLEN) / 256, 256, 0, stream>>>(tags, out);
  finalize<<<1, 256, 0, stream>>>(out, label, seqlen, ll, out);
}